// NNUEModel_10196252361371
// MI455X (gfx1250) — compile-verified
//
#include <hip/hip_runtime.h>

// NNUE model, MI455X (gfx1250, wave32, WMMA)
// Phases: FT sparse gather (BW-bound, float4/b128 path) -> activation/pairwise
// -> W1 GEMM via V_WMMA_F32_16X16X4_F32 with A and all-8-bucket W1 chunks
// staged in LDS (async global->LDS copies, ASYNCcnt-tracked) -> tail MLP.

static constexpr int kL1      = 3072;
static constexpr int kL1P     = 3080;   // L1 + NUM_PSQT
static constexpr int kHalfL1  = 1536;
static constexpr int kBatch   = 1024;
static constexpr int kMaxAct  = 32;
static constexpr int kKC      = 256;          // K-chunk staged in LDS
static constexpr int kAS      = kKC + 4;      // padded row stride (bank skew = 4)

typedef float v2f __attribute__((ext_vector_type(2)));
typedef float v8f __attribute__((ext_vector_type(8)));
typedef int   v4i __attribute__((ext_vector_type(4)));

#define GLOBAL_AS __attribute__((address_space(1)))
#define LDS_AS    __attribute__((address_space(3)))

typedef GLOBAL_AS v4i* gv4p;   // global int4*
typedef LDS_AS    v4i* lv4p;   // LDS int4*

#if __has_builtin(__builtin_amdgcn_global_load_async_to_lds_b128)
#define USE_ASYNC_LDS_CP 1
#endif

// ---------------- Phase 1: feature-transformer sparse accumulation ----------
// Row length 3080 floats (divisible by 4), row base 12320 B (16B aligned):
// full float4 path -> global_load_b128 coalesced across 256 lanes.
__global__ void __launch_bounds__(256)
ft_accum_kernel(const int* __restrict__ wi, const float* __restrict__ wv,
                const int* __restrict__ bi, const float* __restrict__ bv,
                const float* __restrict__ ftw, const float* __restrict__ ftb,
                float* __restrict__ wacc, float* __restrict__ bacc) {
  const int b     = blockIdx.x;
  const int color = blockIdx.y;
  const int*   idx = (color == 0 ? wi : bi) + b * kMaxAct;
  const float* val = (color == 0 ? wv : bv) + b * kMaxAct;
  float* out = (color == 0 ? wacc : bacc) + (size_t)b * kL1P;

  __shared__ int   sidx[kMaxAct];
  __shared__ float sval[kMaxAct];
  if (threadIdx.x < kMaxAct) {
    sidx[threadIdx.x] = idx[threadIdx.x];
    sval[threadIdx.x] = val[threadIdx.x];
  }
  __syncthreads();

  const int nQuads = kL1P / 4;  // 770
  for (int q = threadIdx.x; q < nQuads; q += blockDim.x) {
    float4 acc = *(const float4*)&ftb[q * 4];
#pragma unroll 8
    for (int k = 0; k < kMaxAct; ++k) {
      const float4 r = *(const float4*)&ftw[(size_t)sidx[k] * kL1P + q * 4];
      const float  v = sval[k];
      acc.x += r.x * v; acc.y += r.y * v; acc.z += r.z * v; acc.w += r.w * v;
    }
    *(float4*)&out[q * 4] = acc;
  }
}

// ---------------- Phase 2: clipped-relu + pairwise product -> l0 ------------
__global__ void __launch_bounds__(256)
act_l0_kernel(const float* __restrict__ us, const float* __restrict__ them,
              const float* __restrict__ wacc, const float* __restrict__ bacc,
              float* __restrict__ l0) {
  const int b = blockIdx.x;
  const float u = us[b], t = them[b];
  const float* w  = wacc + (size_t)b * kL1P;
  const float* bk = bacc + (size_t)b * kL1P;
  float* o = l0 + (size_t)b * kL1;
  const float s = 127.0f / 128.0f;

  const int nQuads = kHalfL1 / 4;  // 384
  for (int q = threadIdx.x; q < nQuads; q += blockDim.x) {
    const int j = q * 4;
    float4 w0 = *(const float4*)&w[j];
    float4 w1 = *(const float4*)&w[j + kHalfL1];
    float4 b0 = *(const float4*)&bk[j];
    float4 b1 = *(const float4*)&bk[j + kHalfL1];
    float4 oa, ob;
#pragma unroll
    for (int e = 0; e < 4; ++e) {
      float we0 = (&w0.x)[e], we1 = (&w1.x)[e];
      float be0 = (&b0.x)[e], be1 = (&b1.x)[e];
      float c0 = fminf(fmaxf(u * we0 + t * be0, 0.f), 1.f);
      float c1 = fminf(fmaxf(u * we1 + t * be1, 0.f), 1.f);
      float c2 = fminf(fmaxf(u * be0 + t * we0, 0.f), 1.f);
      float c3 = fminf(fmaxf(u * be1 + t * we1, 0.f), 1.f);
      (&oa.x)[e] = c0 * c1 * s;
      (&ob.x)[e] = c2 * c3 * s;
    }
    *(float4*)&o[j]           = oa;
    *(float4*)&o[j + kHalfL1] = ob;
  }
}

// Helper: one 16B global->LDS transfer (async when the builtin exists).
__device__ __forceinline__ void cp_g2l_b128(const float* gsrc, float* ldst) {
#ifdef USE_ASYNC_LDS_CP
  __builtin_amdgcn_global_load_async_to_lds_b128(
      (gv4p)(GLOBAL_AS void*)gsrc, (lv4p)(LDS_AS void*)ldst, 0, 0);
#else
  *(float4*)ldst = *(const float4*)gsrc;
#endif
}

__device__ __forceinline__ void cp_g2l_wait() {
#ifdef USE_ASYNC_LDS_CP
#if __has_builtin(__builtin_amdgcn_s_wait_asynccnt)
  __builtin_amdgcn_s_wait_asynccnt(0);
#else
  asm volatile("s_wait_asynccnt 0x0" ::: "memory");
#endif
#endif
}

// ---------------- Phase 3: W1 layer via V_WMMA_F32_16X16X4_F32 --------------
// Block = 256 threads = 8 waves; wave w computes bucket w for a 16-row batch
// tile. Per K-chunk: async (ASYNCcnt-tracked) coalesced b128 staging of A
// (16 x KC of l0) AND of W1 for all 8 buckets (128 rows x KC) into LDS with
// padded stride (kAS = KC+4 -> 4-bank row skew, conflict-free ds_load_b64).
// Inner loop grouped 4 WMMAs per ds-load batch to amortize waits.
__global__ void __launch_bounds__(256)
l1_wmma_kernel(const float* __restrict__ l0,
               const float* __restrict__ W1,
               const float* __restrict__ b1,
               const int* __restrict__ lsi,
               float* __restrict__ l1sel) {
  extern __shared__ float smem[];
  float* ldsA = smem;                 // 16  * kAS floats
  float* ldsB = smem + 16 * kAS;      // 128 * kAS floats

  const int tid    = threadIdx.x;
  const int bucket = tid >> 5;        // wave id == bucket (0..7)
  const int lane   = tid & 31;
  const int half   = lane >> 4;       // 0 / 1
  const int mn     = lane & 15;       // A row (M) / B col (N)
  const int batchBase = blockIdx.x * 16;

  const int aoff = mn * kAS + 2 * half;
  const int boff = (bucket * 16 + mn) * kAS + 2 * half;

  v8f acc = {};

  for (int c = 0; c < kL1 / kKC; ++c) {
    const int kb = c * kKC;
    __syncthreads();
    // Stage A chunk: ldsA[m*kAS + kk] = l0[(batchBase+m)*L1 + kb + kk]
#pragma unroll
    for (int i = 0; i < (16 * kKC / 4) / 256; ++i) {      // 4 x 16B / thread
      int q4 = tid + 256 * i;
      int m  = q4 / (kKC / 4);
      int kk = (q4 % (kKC / 4)) * 4;
      cp_g2l_b128(&l0[(size_t)(batchBase + m) * kL1 + kb + kk],
                  &ldsA[m * kAS + kk]);
    }
    // Stage W1 chunk for all 8 buckets: rows 0..127 = bucket*16 + n
#pragma unroll
    for (int i = 0; i < (128 * kKC / 4) / 256; ++i) {     // 32 x 16B / thread
      int q4  = tid + 256 * i;
      int row = q4 / (kKC / 4);
      int kk  = (q4 % (kKC / 4)) * 4;
      cp_g2l_b128(&W1[(size_t)row * kL1 + kb + kk],
                  &ldsB[row * kAS + kk]);
    }
    cp_g2l_wait();
    __syncthreads();

    for (int ks = 0; ks < kKC; ks += 16) {
      v2f a0 = *(const v2f*)&ldsA[aoff + ks];
      v2f a1 = *(const v2f*)&ldsA[aoff + ks + 4];
      v2f a2 = *(const v2f*)&ldsA[aoff + ks + 8];
      v2f a3 = *(const v2f*)&ldsA[aoff + ks + 12];
      v2f b0 = *(const v2f*)&ldsB[boff + ks];
      v2f b1f = *(const v2f*)&ldsB[boff + ks + 4];
      v2f b2 = *(const v2f*)&ldsB[boff + ks + 8];
      v2f b3f = *(const v2f*)&ldsB[boff + ks + 12];
      acc = __builtin_amdgcn_wmma_f32_16x16x4_f32(false, a0, false, b0,  (short)0, acc, false, false);
      acc = __builtin_amdgcn_wmma_f32_16x16x4_f32(false, a1, false, b1f, (short)0, acc, false, false);
      acc = __builtin_amdgcn_wmma_f32_16x16x4_f32(false, a2, false, b2,  (short)0, acc, false, false);
      acc = __builtin_amdgcn_wmma_f32_16x16x4_f32(false, a3, false, b3f, (short)0, acc, false, false);
    }
  }

  // C/D layout: VGPR r -> row M = r + 8*half, col N = mn. Select bucket, add bias.
#pragma unroll
  for (int r = 0; r < 8; ++r) {
    int m = r + 8 * half;
    int batch = batchBase + m;
    if (lsi[batch] == bucket)
      l1sel[batch * 16 + mn] = acc[r] + b1[bucket * 16 + mn];
  }
}

// ---------------- Phase 4: tail MLP (W2, W3) + psqt ------------------------
__global__ void __launch_bounds__(256)
tail_kernel(const float* __restrict__ us, const int* __restrict__ pidx,
            const int* __restrict__ lsi,
            const float* __restrict__ wacc, const float* __restrict__ bacc,
            const float* __restrict__ l1sel,
            const float* __restrict__ W2, const float* __restrict__ b2,
            const float* __restrict__ W3, const float* __restrict__ b3,
            float* __restrict__ out) {
  int b = blockIdx.x * blockDim.x + threadIdx.x;
  if (b >= kBatch) return;
  int li = lsi[b];

  float x[16];
#pragma unroll
  for (int i = 0; i < 16; ++i) x[i] = l1sel[b * 16 + i];
  float l1y = x[15];

  float l1x[30];
#pragma unroll
  for (int i = 0; i < 15; ++i) {
    float v = x[i];
    l1x[i]      = fminf(v * v, 1.f);
    l1x[15 + i] = fminf(fmaxf(v, 0.f), 1.f);
  }

  const float* w2 = W2 + li * 32 * 30;
  float l2x[32];
#pragma unroll
  for (int o = 0; o < 32; ++o) {
    float a = b2[li * 32 + o];
#pragma unroll
    for (int i = 0; i < 30; ++i) a += l1x[i] * w2[o * 30 + i];
    l2x[o] = fminf(fmaxf(a, 0.f), 1.f);
  }

  const float* w3 = W3 + li * 32;
  float l3 = b3[li];
#pragma unroll
  for (int o = 0; o < 32; ++o) l3 += l2x[o] * w3[o];

  int pi = pidx[b];
  float wps = wacc[(size_t)b * kL1P + kL1 + pi];
  float bps = bacc[(size_t)b * kL1P + kL1 + pi];
  out[b] = l3 + l1y + (wps - bps) * (us[b] - 0.5f);
}

// ---------------------------------------------------------------------------
extern "C" void kernel_launch(void* const* d_in, const int* in_sizes, int n_in,
                              void* d_out, int out_size, void* d_ws, size_t ws_size,
                              hipStream_t stream) {
  const float* us   = (const float*)d_in[0];
  const float* them = (const float*)d_in[1];
  const int*   wi   = (const int*)  d_in[2];
  const float* wv   = (const float*)d_in[3];
  const int*   bi   = (const int*)  d_in[4];
  const float* bv   = (const float*)d_in[5];
  const int*   pidx = (const int*)  d_in[6];
  const int*   lsi  = (const int*)  d_in[7];
  const float* ftw  = (const float*)d_in[8];
  const float* ftb  = (const float*)d_in[9];
  const float* W1   = (const float*)d_in[10];
  const float* b1   = (const float*)d_in[11];
  const float* W2   = (const float*)d_in[12];
  const float* b2   = (const float*)d_in[13];
  const float* W3   = (const float*)d_in[14];
  const float* b3   = (const float*)d_in[15];
  float* out = (float*)d_out;

  float* ws    = (float*)d_ws;
  float* wacc  = ws;                                     // B * 3080
  float* bacc  = wacc  + (size_t)kBatch * kL1P;          // B * 3080
  float* l0    = bacc  + (size_t)kBatch * kL1P;          // B * 3072
  float* l1sel = l0    + (size_t)kBatch * kL1;           // B * 16

  ft_accum_kernel<<<dim3(kBatch, 2), 256, 0, stream>>>(wi, wv, bi, bv, ftw, ftb,
                                                       wacc, bacc);
  act_l0_kernel<<<kBatch, 256, 0, stream>>>(us, them, wacc, bacc, l0);

  const size_t ldsBytes = (size_t)(16 + 128) * kAS * sizeof(float);  // ~150 KB
  l1_wmma_kernel<<<kBatch / 16, 256, ldsBytes, stream>>>(l0, W1, b1, lsi, l1sel);

  tail_kernel<<<(kBatch + 255) / 256, 256, 0, stream>>>(
      us, pidx, lsi, wacc, bacc, l1sel, W2, b2, W3, b3, out);
}